// AtssLoss_49847390437862
// MI455X (gfx1250) — compile-verified
//
#include <hip/hip_runtime.h>
#include <math.h>

#define A_TOTAL 22400
#define NUM_G   32
#define NUM_C   80
#define TILE    128   // anchors per block in loss kernel (A_TOTAL = 175*128)

#if defined(__gfx1250__) && __has_builtin(__builtin_amdgcn_wmma_f32_16x16x4_f32)
#define USE_WMMA4 1
#else
#define USE_WMMA4 0
#endif

typedef float v2f __attribute__((ext_vector_type(2)));
typedef float v8f __attribute__((ext_vector_type(8)));

__device__ __forceinline__ float softplusf_(float z) {
  // log1p(exp(z)) stable
  return fmaxf(z, 0.0f) + log1pf(expf(-fabsf(z)));
}

// focal loss element: -(t==c)*alpha*(1-p)^2*logsig(x) - (t!=c)*(1-alpha)*p^2*logsig(-x)
__device__ __forceinline__ float focal_term(float x, int t, int c) {
  float p = 1.0f / (1.0f + expf(-x));
  if (t == c) { float q = 1.0f - p; return 0.25f * q * q * softplusf_(-x); }
  return 0.75f * p * p * softplusf_(x);
}

__device__ __forceinline__ float block_sum256(float v, float* s_red) {
  int tid = threadIdx.x;
  s_red[tid] = v;
  __syncthreads();
  for (int off = 128; off > 0; off >>= 1) {
    if (tid < off) s_red[tid] += s_red[tid + off];
    __syncthreads();
  }
  float r = s_red[0];
  __syncthreads();
  return r;
}

// ---------------------------------------------------------------------------
// Kernel 1: ATSS assignment. One block per (b, g).
// Writes pos_key[b*A + anchor] = ((iou_bits+1) << 32) | (31 - g) via atomicMax.
// ---------------------------------------------------------------------------
__global__ void atss_assign(const float* __restrict__ gt_bboxes,
                            const float* __restrict__ anchors,
                            unsigned long long* __restrict__ pos_key) {
  const int b = blockIdx.x >> 5;   // / 32
  const int g = blockIdx.x & 31;
  const int tid = threadIdx.x;

  __shared__ unsigned long long s_min;
  __shared__ int   s_cand[45];
  __shared__ float s_ov[45];
  __shared__ float s_thr;

  const int LVL_START[5] = {0, 16800, 21000, 22050, 22323};
  const int LVL_SIZE[5]  = {16800, 4200, 1050, 273, 77};

  const float* gtb = gt_bboxes + ((size_t)b * NUM_G + g) * 4;
  const float gx1 = gtb[0], gy1 = gtb[1], gx2 = gtb[2], gy2 = gtb[3];
  const float gcx = (gx1 + gx2) * 0.5f, gcy = (gy1 + gy2) * 0.5f;

  // Per level: 9 passes of block-wide min over packed (dist2, idx).
  for (int l = 0; l < 5; ++l) {
    const int start = LVL_START[l], n = LVL_SIZE[l];
    for (int p = 0; p < 9; ++p) {
      if (tid == 0) s_min = ~0ULL;
      __syncthreads();
      unsigned long long local = ~0ULL;
      for (int i = start + tid; i < start + n; i += (int)blockDim.x) {
        bool used = false;
        for (int q = 0; q < p; ++q) used |= (s_cand[l * 9 + q] == i);
        if (used) continue;
        float ax1 = anchors[4 * i + 0], ay1 = anchors[4 * i + 1];
        float ax2 = anchors[4 * i + 2], ay2 = anchors[4 * i + 3];
        float dx = (ax1 + ax2) * 0.5f - gcx;
        float dy = (ay1 + ay2) * 0.5f - gcy;
        float d2 = dx * dx + dy * dy;   // monotonic w.r.t. sqrt -> same ranking
        unsigned long long key =
            ((unsigned long long)__float_as_uint(d2) << 32) | (unsigned int)i;
        if (key < local) local = key;
      }
      atomicMin(&s_min, local);
      __syncthreads();
      if (tid == 0) s_cand[l * 9 + p] = (int)(s_min & 0xffffffffu);
      __syncthreads();
    }
  }

  // IoU of the 45 candidates with this gt
  const float garea = (gx2 - gx1) * (gy2 - gy1);
  if (tid < 45) {
    int i = s_cand[tid];
    float ax1 = anchors[4 * i], ay1 = anchors[4 * i + 1];
    float ax2 = anchors[4 * i + 2], ay2 = anchors[4 * i + 3];
    float iw = fmaxf(fminf(ax2, gx2) - fmaxf(ax1, gx1), 0.0f);
    float ih = fmaxf(fminf(ay2, gy2) - fmaxf(ay1, gy1), 0.0f);
    float inter = iw * ih;
    float aarea = (ax2 - ax1) * (ay2 - ay1);
    s_ov[tid] = inter / (aarea + garea - inter + 1e-6f);
  }
  __syncthreads();
  if (tid == 0) {
    float m = 0.0f;
    for (int i = 0; i < 45; ++i) m += s_ov[i];
    m *= (1.0f / 45.0f);
    float v = 0.0f;
    for (int i = 0; i < 45; ++i) { float d = s_ov[i] - m; v += d * d; }
    v *= (1.0f / 44.0f);               // ddof=1
    s_thr = m + sqrtf(v);
  }
  __syncthreads();
  if (tid < 45) {
    float ov = s_ov[tid];
    int i = s_cand[tid];
    float ax1 = anchors[4 * i], ay1 = anchors[4 * i + 1];
    float ax2 = anchors[4 * i + 2], ay2 = anchors[4 * i + 3];
    float acx = (ax1 + ax2) * 0.5f, acy = (ay1 + ay2) * 0.5f;
    float le = acx - gx1, te = acy - gy1, re = gx2 - acx, be = gy2 - acy;
    float ins = fminf(fminf(le, re), fminf(te, be));
    if (ov >= s_thr && ins > 0.01f) {
      unsigned long long key =
          ((unsigned long long)(__float_as_uint(ov) + 1u) << 32) |
          (unsigned long long)(31 - g);          // tie-break: smaller g wins argmax
      atomicMax(&pos_key[(size_t)b * A_TOTAL + i], key);
    }
  }
}

// ---------------------------------------------------------------------------
// Kernel 2: main loss. Block = 256 threads = 8 waves, 128 anchors per block.
// Waves do the 80-class focal reduction with V_WMMA_F32_16X16X4_F32.
// Threads 0..127 do decode/GIoU/centerness for their anchor.
// ---------------------------------------------------------------------------
__global__ void atss_loss(const float* __restrict__ cls,
                          const float* __restrict__ reg,
                          const float* __restrict__ cen,
                          const float* __restrict__ gt_bboxes,
                          const int*   __restrict__ gt_labels,
                          const float* __restrict__ anchors,
                          const unsigned long long* __restrict__ pos_key,
                          float* __restrict__ partials) {
  const int b    = blockIdx.y;
  const int tile = blockIdx.x * TILE;
  const int tid  = threadIdx.x;

  __shared__ float s_red[256];
  __shared__ float s_wave[8];

  // ---------------- per-anchor regression / centerness (threads 0..127) ----
  float np_l = 0.0f, cw_l = 0.0f, rg_l = 0.0f, cn_l = 0.0f;
  if (tid < TILE) {
    int a = tile + tid;
    unsigned long long key = pos_key[(size_t)b * A_TOTAL + a];
    if (key != 0ULL) {
      int g = 31 - (int)(key & 0xffffffffu);
      const float* gtb = gt_bboxes + ((size_t)b * NUM_G + g) * 4;
      float gx1 = gtb[0], gy1 = gtb[1], gx2 = gtb[2], gy2 = gtb[3];
      float ax1 = anchors[4 * a], ay1 = anchors[4 * a + 1];
      float ax2 = anchors[4 * a + 2], ay2 = anchors[4 * a + 3];
      float aw = ax2 - ax1, ah = ay2 - ay1;
      float acx = (ax1 + ax2) * 0.5f, acy = (ay1 + ay2) * 0.5f;
      // centerness target
      float le = acx - gx1, te = acy - gy1, re = gx2 - acx, be = gy2 - acy;
      float ratio = (fminf(le, re) / fmaxf(le, re)) * (fminf(te, be) / fmaxf(te, be));
      float cen_t = sqrtf(fmaxf(ratio, 1e-12f));
      // decode
      const float* rp = reg + ((size_t)b * A_TOTAL + a) * 4;
      float dx = rp[0] * 0.1f, dy = rp[1] * 0.1f;
      float dw = fminf(fmaxf(rp[2] * 0.2f, -4.135f), 4.135f);
      float dh = fminf(fmaxf(rp[3] * 0.2f, -4.135f), 4.135f);
      float px = acx + dx * aw, py = acy + dy * ah;
      float pw = aw * expf(dw), ph = ah * expf(dh);
      float p1 = px - pw * 0.5f, p2 = py - ph * 0.5f;
      float p3 = px + pw * 0.5f, p4 = py + ph * 0.5f;
      // GIoU loss
      float iw = fmaxf(fminf(p3, gx2) - fmaxf(p1, gx1), 0.0f);
      float ih = fmaxf(fminf(p4, gy2) - fmaxf(p2, gy1), 0.0f);
      float ov = iw * ih;
      float ap = (p3 - p1) * (p4 - p2);
      float ag = (gx2 - gx1) * (gy2 - gy1);
      float uni = ap + ag - ov + 1.0f;
      float iou = ov / uni;
      float ew = fmaxf(fmaxf(p3, gx2) - fminf(p1, gx1), 0.0f);
      float eh = fmaxf(fmaxf(p4, gy2) - fminf(p2, gy1), 0.0f);
      float ea = ew * eh + 1.0f;
      float gl = 1.0f - (iou - (ea - uni) / ea);
      // centerness BCE
      float x = cen[(size_t)b * A_TOTAL + a];
      float bce = fmaxf(x, 0.0f) - x * cen_t + log1pf(expf(-fabsf(x)));
      np_l = 1.0f; cw_l = cen_t; rg_l = gl * cen_t; cn_l = bce;
    }
  }

  // ---------------- focal loss over 80 classes, 16 anchors per wave --------
  const int lane = tid & 31, wave = tid >> 5;
  const int m = lane & 15, half = lane >> 4;   // half: which pair of K slots
  const int a = tile + wave * 16 + m;
  unsigned long long key = pos_key[(size_t)b * A_TOTAL + a];
  int label = NUM_C;
  if (key != 0ULL) {
    int g = 31 - (int)(key & 0xffffffffu);
    label = gt_labels[b * NUM_G + g];
  }
  const float* crow = cls + ((size_t)b * A_TOTAL + a) * NUM_C;
  __builtin_prefetch(crow, 0, 3);              // global_prefetch_b8

  float cls_partial;                           // sum over this wave's 16x80 tile
#if USE_WMMA4
  // D = A(16x4) * ones(4x16) + C  ==> D[m][n] = sum_k A[m][k] + C[m][n]
  // A layout (ISA 7.12.2): lanes 0-15 hold K=0,1 ; lanes 16-31 hold K=2,3.
  v8f acc = {0.f, 0.f, 0.f, 0.f, 0.f, 0.f, 0.f, 0.f};
  v2f bones = {1.0f, 1.0f};
  for (int s = 0; s < 20; ++s) {
    int c0 = 4 * s + 2 * half;
    float t0 = focal_term(crow[c0],     label, c0);
    float t1 = focal_term(crow[c0 + 1], label, c0 + 1);
    v2f aop = {t0, t1};
    acc = __builtin_amdgcn_wmma_f32_16x16x4_f32(
        false, aop, false, bones, (short)0, acc, false, false);
  }
  // every column identical: lane 0 holds rows 0..7, lane 16 holds rows 8..15
  float s8 = acc[0] + acc[1] + acc[2] + acc[3] + acc[4] + acc[5] + acc[6] + acc[7];
  cls_partial = __shfl(s8, 0, 32) + __shfl(s8, 16, 32);
#else
  float accs = 0.0f;
  for (int s = 0; s < 20; ++s) {
    int c0 = 4 * s + 2 * half;
    accs += focal_term(crow[c0],     label, c0);
    accs += focal_term(crow[c0 + 1], label, c0 + 1);
  }
  for (int off = 16; off > 0; off >>= 1) accs += __shfl_xor(accs, off, 32);
  cls_partial = accs;
#endif
  if (lane == 0) s_wave[wave] = cls_partial;
  __syncthreads();
  float cls_block = 0.0f;
  if (tid == 0)
    for (int w = 0; w < 8; ++w) cls_block += s_wave[w];   // fixed order

  // ---------------- deterministic block reductions & output ----------------
  float np_b = block_sum256(np_l, s_red);
  float cw_b = block_sum256(cw_l, s_red);
  float rg_b = block_sum256(rg_l, s_red);
  float cn_b = block_sum256(cn_l, s_red);
  if (tid == 0) {
    int blk = blockIdx.y * gridDim.x + blockIdx.x;
    float* o = partials + (size_t)blk * 5;
    o[0] = np_b; o[1] = cw_b; o[2] = rg_b; o[3] = cn_b; o[4] = cls_block;
  }
}

// ---------------------------------------------------------------------------
// Kernel 3: final reduction + loss combination. One block.
// ---------------------------------------------------------------------------
__global__ void atss_finalize(const float* __restrict__ partials, int nblk,
                              float* __restrict__ out) {
  __shared__ float s_red[256];
  int tid = threadIdx.x;
  float v[5] = {0.f, 0.f, 0.f, 0.f, 0.f};
  for (int i = tid; i < nblk; i += 256)
    for (int j = 0; j < 5; ++j) v[j] += partials[(size_t)i * 5 + j];
  float tot[5];
  for (int j = 0; j < 5; ++j) {
    s_red[tid] = v[j];
    __syncthreads();
    for (int off = 128; off > 0; off >>= 1) {
      if (tid < off) s_red[tid] += s_red[tid + off];
      __syncthreads();
    }
    tot[j] = s_red[0];
    __syncthreads();
  }
  if (tid == 0) {
    float num_pos  = fmaxf(tot[0], 1.0f);
    float cls_loss = tot[4] / num_pos;                       // CLS_W = 1
    float reg_loss = 2.0f * tot[2] / fmaxf(tot[1], 1e-6f);   // REG_W = 2
    float cen_loss = tot[3] / num_pos;                       // CEN_W = 1
    out[0] = cls_loss + reg_loss + cen_loss;
  }
}

// ---------------------------------------------------------------------------
extern "C" void kernel_launch(void* const* d_in, const int* in_sizes, int n_in,
                              void* d_out, int out_size, void* d_ws, size_t ws_size,
                              hipStream_t stream) {
  const float* cls     = (const float*)d_in[0];
  const float* reg     = (const float*)d_in[1];
  const float* cen     = (const float*)d_in[2];
  const float* gtb     = (const float*)d_in[3];
  const int*   gtl     = (const int*)d_in[4];
  const float* anchors = (const float*)d_in[5];

  const int B = in_sizes[2] / A_TOTAL;            // cen_preds is (B, A_TOTAL)
  const int blocks_per_img = A_TOTAL / TILE;      // 175

  unsigned long long* pos_key = (unsigned long long*)d_ws;
  const size_t pk_bytes = (size_t)B * A_TOTAL * sizeof(unsigned long long);
  float* partials = (float*)((char*)d_ws + pk_bytes);

  hipMemsetAsync(pos_key, 0, pk_bytes, stream);

  atss_assign<<<dim3(B * NUM_G), dim3(256), 0, stream>>>(gtb, anchors, pos_key);
  atss_loss<<<dim3(blocks_per_img, B), dim3(256), 0, stream>>>(
      cls, reg, cen, gtb, gtl, anchors, pos_key, partials);
  atss_finalize<<<dim3(1), dim3(256), 0, stream>>>(
      partials, B * blocks_per_img, (float*)d_out);
}